// MultiBoxLoss_1769526526491
// MI455X (gfx1250) — compile-verified
//
#include <hip/hip_runtime.h>
#include <hip/hip_bf16.h>

// ---- problem constants (match reference) ----
constexpr int NB = 32;       // batch
constexpr int NP = 19248;    // priors
constexpr int NG = 32;       // gt boxes per image
constexpr int NC = 41;       // classes (40 fg + background)
constexpr float POS_TH = 0.5f, NEG_TH = 0.4f;
constexpr float BBOX_ALPHA = 1.5f, CONF_ALPHA = 6.5f;
constexpr float VAR_C = 0.1f, VAR_S = 0.2f;

typedef __attribute__((ext_vector_type(2))) float v2f;
typedef __attribute__((ext_vector_type(8))) float v8f;

// ---------------- Kernel 1: matching + box loss (one block per batch) ------
__global__ __launch_bounds__(256) void k_match(
    const float* __restrict__ loc, const float* __restrict__ priors,
    const float* __restrict__ gtb, const int* __restrict__ gtl,
    float* __restrict__ best_ov, int* __restrict__ best_idx,
    int* __restrict__ conf_t, int* __restrict__ pos_cnt, float* __restrict__ box_l)
{
  const int b = blockIdx.x, tid = threadIdx.x;
  __shared__ float4 sg[NG];
  __shared__ int    slab[NG];
  __shared__ unsigned long long skey[NG];  // (iou_bits<<32) | (~p) -> argmax w/ low-index ties
  __shared__ int    sfp[NG];               // best prior per gt
  __shared__ float  srf[256];
  __shared__ int    sri[256];

  if (tid < NG) {
    size_t o = ((size_t)b * NG + tid) * 4;
    sg[tid]   = make_float4(gtb[o+0], gtb[o+1], gtb[o+2], gtb[o+3]);
    slab[tid] = gtl[(size_t)b * NG + tid];
    skey[tid] = 0ull;
  }
  __syncthreads();

  for (int p = tid; p < NP; p += 256) {
    float cx = priors[p*4+0], cy = priors[p*4+1], w = priors[p*4+2], h = priors[p*4+3];
    float bx1 = cx - 0.5f*w, by1 = cy - 0.5f*h, bx2 = cx + 0.5f*w, by2 = cy + 0.5f*h;
    float areab = w * h;
    float best = -1.0f; int bi = 0;
    #pragma unroll 8
    for (int g = 0; g < NG; ++g) {
      float4 a = sg[g];
      float iw = fminf(a.z, bx2) - fmaxf(a.x, bx1);
      float ih = fminf(a.w, by2) - fmaxf(a.y, by1);
      float inter = fmaxf(iw, 0.f) * fmaxf(ih, 0.f);
      float areaa = (a.z - a.x) * (a.w - a.y);
      float iou = inter / (areaa + areab - inter);
      if (iou > best) { best = iou; bi = g; }   // strict > : first-max tie-break (JAX argmax)
      unsigned long long key =
          ((unsigned long long)__float_as_uint(iou) << 32) |
          (unsigned long long)(0xFFFFFFFFu - (unsigned)p);
      atomicMax(&skey[g], key);
    }
    best_ov[(size_t)b*NP + p]  = best;
    best_idx[(size_t)b*NP + p] = bi;
  }
  __syncthreads();
  if (tid < NG) sfp[tid] = (int)(0xFFFFFFFFu - (unsigned)(skey[tid] & 0xFFFFFFFFull));
  __syncthreads();

  float accB = 0.f; int accP = 0;
  for (int p = tid; p < NP; p += 256) {
    float bo = best_ov[(size_t)b*NP + p];
    int   bi = best_idx[(size_t)b*NP + p];
    #pragma unroll
    for (int g = 0; g < NG; ++g)                 // forced matches, last-wins in g order
      if (sfp[g] == p) { bo = 2.0f; bi = g; }
    float4 m = sg[bi];
    int ct = slab[bi] + 1;
    if (bo < POS_TH) ct = -1;
    if (bo < NEG_TH) ct = 0;
    conf_t[(size_t)b*NP + p] = ct;

    float cx = priors[p*4+0], cy = priors[p*4+1], w = priors[p*4+2], h = priors[p*4+3];
    float t0 = ((m.x + m.z)*0.5f - cx) / (VAR_C * w);
    float t1 = ((m.y + m.w)*0.5f - cy) / (VAR_C * h);
    float t2 = logf(fmaxf((m.z - m.x) / w, 1e-8f)) / VAR_S;
    float t3 = logf(fmaxf((m.w - m.y) / h, 1e-8f)) / VAR_S;
    const float* lp = loc + ((size_t)b*NP + p) * 4;
    float s = 0.f, d;
    d = fabsf(lp[0]-t0); s += (d < 1.f) ? 0.5f*d*d : d - 0.5f;
    d = fabsf(lp[1]-t1); s += (d < 1.f) ? 0.5f*d*d : d - 0.5f;
    d = fabsf(lp[2]-t2); s += (d < 1.f) ? 0.5f*d*d : d - 0.5f;
    d = fabsf(lp[3]-t3); s += (d < 1.f) ? 0.5f*d*d : d - 0.5f;
    if (ct > 0) { accP += 1; accB += s; }
  }
  srf[tid] = accB; sri[tid] = accP;
  __syncthreads();
  for (int s = 128; s > 0; s >>= 1) {
    if (tid < s) { srf[tid] += srf[tid+s]; sri[tid] += sri[tid+s]; }
    __syncthreads();
  }
  if (tid == 0) { pos_cnt[b] = sri[0]; box_l[b] = srf[0]; }
}

// ---------------- Kernel 2: per-prior CE loss, sumexp via WMMA -------------
__device__ __forceinline__ float sel8(const float x[8], int r) {
  float s = x[0];
  #pragma unroll
  for (int j = 1; j < 8; ++j) s = (r == j) ? x[j] : s;
  return s;
}

__global__ __launch_bounds__(256) void k_conf(
    const float* __restrict__ conf, const int* __restrict__ conf_t,
    float* __restrict__ loss_c)
{
  const int b    = blockIdx.y;
  const int lane = threadIdx.x & 31;
  const int wave = threadIdx.x >> 5;
  const int pl   = lane & 15;                     // prior-in-wave == WMMA row M
  const int p    = blockIdx.x * 128 + wave * 16 + pl;
  const int pc   = (p < NP) ? p : (NP - 1);       // clamp: keep EXEC all-ones for WMMA
  const float* row = conf + ((size_t)b * NP + pc) * NC;
  __builtin_prefetch(row, 0, 1);                  // global_prefetch_b8

  // stable max: half-waves take even/odd classes, merge across the xor-16 pair
  float mx = -3.4e38f;
  for (int c = (lane >> 4); c < NC; c += 2) mx = fmaxf(mx, row[c]);
  mx = fmaxf(mx, __shfl_xor(mx, 16, 32));

  // Sum_k exp(l_k - mx) as D = A x ones + C with V_WMMA_F32_16X16X4_F32.
  // A layout (16x4 f32): lanes 0-15 -> K=0,1 ; lanes 16-31 -> K=2,3 ; M = lane&15.
  v8f acc = {};
  float srow = 0.f;
#if __has_builtin(__builtin_amdgcn_wmma_f32_16x16x4_f32)
  v2f ones = {1.f, 1.f};
  #pragma unroll
  for (int k0 = 0; k0 < 44; k0 += 4) {
    int c0 = k0 + ((lane >> 4) << 1);
    float e0 = (c0     < NC) ? expf(row[c0]     - mx) : 0.f;
    float e1 = (c0 + 1 < NC) ? expf(row[c0 + 1] - mx) : 0.f;
    v2f a = {e0, e1};
    acc = __builtin_amdgcn_wmma_f32_16x16x4_f32(
        false, a, false, ones, (short)0, acc, false, false);
  }
  // D rows 0-7 live in lanes 0-15 (vgpr j = row), rows 8-15 in lanes 16-31.
  float tmp[8], oth[8];
  #pragma unroll
  for (int j = 0; j < 8; ++j) tmp[j] = acc[j];
  #pragma unroll
  for (int j = 0; j < 8; ++j) oth[j] = __shfl_xor(tmp[j], 16, 32);
  srow = (pl < 8) ? sel8(tmp, pl) : sel8(oth, pl - 8);
#else
  (void)acc;
  for (int c = 0; c < NC; ++c) srow += expf(row[c] - mx);
#endif

  if (lane < 16 && p < NP) {
    float lse = mx + logf(srow);
    int ct = conf_t[(size_t)b * NP + p];
    int tg = (ct > 0) ? ct : 0;                   // clip(conf_t, 0)
    loss_c[(size_t)b * NP + p] = lse - row[tg];
  }
}

// ---------------- Kernel 3: per-batch two-level radix top-K threshold ------
constexpr int NBIN = 8192;                        // 13 bits/level, 32KB LDS
__global__ __launch_bounds__(256) void k_select(
    const int* __restrict__ conf_t, const float* __restrict__ loss_c,
    const int* __restrict__ pos_cnt, unsigned* __restrict__ pref_out,
    int* __restrict__ rrem_out)
{
  const int b = blockIdx.x, tid = threadIdx.x;
  __shared__ unsigned hist[NBIN];
  __shared__ int s_t, s_chi, s_s, s_cmid;

  int num_neg = pos_cnt[b] * 3;
  if (num_neg > NP - 1) num_neg = NP - 1;
  if (num_neg <= 0) {                             // uniform early out
    if (tid == 0) { pref_out[b] = 0xFFFFFFFFu; rrem_out[b] = 0; }
    return;
  }
  for (int i = tid; i < NBIN; i += 256) hist[i] = 0u;
  __syncthreads();
  for (int p = tid; p < NP; p += 256) {
    int ct = conf_t[(size_t)b*NP + p];
    float mv = (ct == 0) ? fmaxf(loss_c[(size_t)b*NP + p], 0.f) : 0.f;
    atomicAdd(&hist[__float_as_uint(mv) >> 19], 1u);
  }
  __syncthreads();
  if (tid == 0) {
    unsigned cum = 0;
    for (int i = NBIN - 1; i >= 0; --i) {
      unsigned nc = cum + hist[i];
      if (nc >= (unsigned)num_neg) { s_t = i; s_chi = (int)cum; break; }
      cum = nc;
    }
  }
  __syncthreads();
  const int t = s_t, chi = s_chi;
  for (int i = tid; i < NBIN; i += 256) hist[i] = 0u;
  __syncthreads();
  for (int p = tid; p < NP; p += 256) {
    int ct = conf_t[(size_t)b*NP + p];
    float mv = (ct == 0) ? fmaxf(loss_c[(size_t)b*NP + p], 0.f) : 0.f;
    unsigned bits = __float_as_uint(mv);
    if ((int)(bits >> 19) == t) atomicAdd(&hist[(bits >> 6) & (NBIN - 1)], 1u);
  }
  __syncthreads();
  if (tid == 0) {
    unsigned cum = 0;
    for (int i = NBIN - 1; i >= 0; --i) {
      unsigned nc = cum + hist[i];
      if (chi + (int)nc >= num_neg) { s_s = i; s_cmid = (int)cum; break; }
      cum = nc;
    }
    pref_out[b] = ((unsigned)s_t << 13) | (unsigned)s_s;   // == mined_bits >> 6
    rrem_out[b] = num_neg - chi - s_cmid;
  }
}

// ---------------- Kernel 4: conf-loss accumulation (pos + mined negs) ------
__global__ __launch_bounds__(256) void k_neg(
    const int* __restrict__ conf_t, const float* __restrict__ loss_c,
    const unsigned* __restrict__ pref_arr, const int* __restrict__ rrem_arr,
    float* __restrict__ conf_l)
{
  const int b = blockIdx.x, tid = threadIdx.x;
  const unsigned pref = pref_arr[b];
  const int rrem = rrem_arr[b];
  __shared__ float srf[256];

  float acc = 0.f;
  for (int p = tid; p < NP; p += 256) {
    int ct = conf_t[(size_t)b*NP + p];
    float lc = loss_c[(size_t)b*NP + p];
    if (ct > 0) acc += lc;
    else if (ct == 0) {
      unsigned key = __float_as_uint(fmaxf(lc, 0.f)) >> 6;
      if (key > pref) acc += lc;                  // strictly above threshold group
    }
  }
  // boundary group: wave 0, index-ordered ballot scan (stable tie-break)
  if (tid < 32 && rrem > 0) {
    int taken = 0;
    for (int base = 0; base < NP && taken < rrem; base += 32) {
      int p = base + tid;
      bool eq = false; float lc = 0.f;
      if (p < NP) {
        int ct = conf_t[(size_t)b*NP + p];
        lc = loss_c[(size_t)b*NP + p];
        eq = (ct == 0) && ((__float_as_uint(fmaxf(lc, 0.f)) >> 6) == pref);
      }
      unsigned mask = (unsigned)__ballot(eq);
      int before = __popc(mask & ((1u << tid) - 1u));
      if (eq && taken + before < rrem) acc += lc;
      taken += __popc(mask);
    }
  }
  srf[tid] = acc;
  __syncthreads();
  for (int s = 128; s > 0; s >>= 1) {
    if (tid < s) srf[tid] += srf[tid + s];
    __syncthreads();
  }
  if (tid == 0) conf_l[b] = srf[0];
}

// ---------------- Kernel 5: final combine ----------------------------------
__global__ void k_final(const int* __restrict__ pos_cnt,
                        const float* __restrict__ box_l,
                        const float* __restrict__ conf_l,
                        float* __restrict__ out)
{
  if (threadIdx.x == 0 && blockIdx.x == 0) {
    int tp = 0; float bl = 0.f, cl = 0.f;
    for (int b = 0; b < NB; ++b) { tp += pos_cnt[b]; bl += box_l[b]; cl += conf_l[b]; }
    float np = (float)((tp > 0) ? tp : 1);
    out[0] = bl / np * BBOX_ALPHA + cl / np * CONF_ALPHA;
  }
}

// ---------------- host-side launcher ---------------------------------------
extern "C" void kernel_launch(void* const* d_in, const int* in_sizes, int n_in,
                              void* d_out, int out_size, void* d_ws, size_t ws_size,
                              hipStream_t stream) {
  (void)in_sizes; (void)n_in; (void)out_size; (void)ws_size;
  const float* loc    = (const float*)d_in[0];
  const float* conf   = (const float*)d_in[1];
  const float* priors = (const float*)d_in[2];
  const float* gtb    = (const float*)d_in[3];
  const int*   gtl    = (const int*)d_in[4];
  float* out = (float*)d_out;

  char* w = (char*)d_ws;
  const size_t BP = (size_t)NB * NP;
  float*    ws_best_ov  = (float*)w;    w += BP * 4;
  int*      ws_best_idx = (int*)w;      w += BP * 4;
  int*      ws_conf_t   = (int*)w;      w += BP * 4;
  float*    ws_loss_c   = (float*)w;    w += BP * 4;
  int*      ws_pos      = (int*)w;      w += NB * 4;
  float*    ws_box      = (float*)w;    w += NB * 4;
  float*    ws_conf     = (float*)w;    w += NB * 4;
  unsigned* ws_pref     = (unsigned*)w; w += NB * 4;
  int*      ws_rrem     = (int*)w;      w += NB * 4;

  k_match<<<NB, 256, 0, stream>>>(loc, priors, gtb, gtl,
                                  ws_best_ov, ws_best_idx, ws_conf_t, ws_pos, ws_box);
  dim3 g2((NP + 127) / 128, NB);
  k_conf<<<g2, 256, 0, stream>>>(conf, ws_conf_t, ws_loss_c);
  k_select<<<NB, 256, 0, stream>>>(ws_conf_t, ws_loss_c, ws_pos, ws_pref, ws_rrem);
  k_neg<<<NB, 256, 0, stream>>>(ws_conf_t, ws_loss_c, ws_pref, ws_rrem, ws_conf);
  k_final<<<1, 32, 0, stream>>>(ws_pos, ws_box, ws_conf, out);
}